// BalancedVQVAE_20315195310706
// MI455X (gfx1250) — compile-verified
//
#include <hip/hip_runtime.h>
#include <hip/hip_bf16.h>

#define NQ (16 * 16384)   // 262144 rows
#define DQ 64             // embedding dim
#define KQ 512            // codebook size
#define CBS 68            // LDS codebook row stride in floats (64 + 4 TDM pad)

typedef __attribute__((ext_vector_type(2))) float v2f;
typedef __attribute__((ext_vector_type(8))) float v8f;
typedef unsigned int u32x4 __attribute__((ext_vector_type(4)));
typedef int i32x4 __attribute__((ext_vector_type(4)));
typedef int i32x8 __attribute__((ext_vector_type(8)));

// ---------------- prep: codebook squared norms + zero accumulators ----------
__global__ void vq_prep(const float* __restrict__ emb,
                        float* __restrict__ enorm,
                        int* __restrict__ counts,
                        float* __restrict__ sse) {
  int k = threadIdx.x;  // 512 threads
  float s = 0.f;
#pragma unroll
  for (int d = 0; d < DQ; ++d) {
    float v = emb[k * DQ + d];
    s += v * v;
  }
  enorm[k] = s;
  counts[k] = 0;
  if (k == 0) *sse = 0.f;
}

// ---------------- WMMA distance + argmin + quantized gather -----------------
// 128 threads = 4 wave32; each wave handles 16 rows of X against all 512 codes.
// Codebook staged into LDS once per block by the Tensor Data Mover, with TDM
// padding of 4 DWORDs per 64 DWORDs -> LDS row stride 68 (bank-conflict-free).
__global__ void __launch_bounds__(128)
vq_argmin(const float* __restrict__ X, const float* __restrict__ E,
          const float* __restrict__ enorm, int* __restrict__ idx_g,
          float* __restrict__ out_q, float* __restrict__ out_idxf,
          int* __restrict__ counts, float* __restrict__ sse_g) {
  __shared__ float cb[KQ * CBS];   // 139264 B padded codebook
  __shared__ float ens[KQ];        // code norms
  __shared__ int idx_s[64];
  __shared__ float red_s[128];

  const int tid = threadIdx.x;
  const int lane = tid & 31;
  const int wave = tid >> 5;
  const int row0 = blockIdx.x * 64;     // block's first row of X
  const int wrow0 = row0 + wave * 16;   // this wave's first row

  const int rm = lane & 15;  // A: row in tile; B/C: column in tile
  const int hi = lane >> 4;  // half selector (K-pair for A/B, M+8 for C)

  // ---- TDM: codebook (512x64 f32, 128 KB) -> LDS with padding --------------
  if (wave == 0) {
    const unsigned long long ga = (unsigned long long)(uintptr_t)E;
    const unsigned lds_off = (unsigned)(uintptr_t)(void*)cb;  // addr[31:0] = LDS byte offset
    const unsigned ne = KQ * DQ;  // 32768 elements of 4 bytes
    // D# group 0: count=1 | lds_addr | global_addr[56:0] | type=2
    u32x4 g0 = {1u, lds_off, (unsigned)ga,
                (unsigned)((ga >> 32) & 0x01FFFFFFull) | 0x80000000u};
    // D# group 1: data_size=4B, pad_enable, pad_interval=64 DW, pad_amount=4 DW
    //             tensor_dim0 = tile_dim0 = 32768, tensor_dim1 = tile_dim1 = 1
    i32x8 g1 = {
        (int)((2u << 16) | (1u << 20) | (5u << 22) | (3u << 25)),  // dw0
        (int)((ne & 0xFFFFu) << 16),                               // dw1: dim0 lo
        (int)(((ne >> 16) & 0xFFFFu) | (1u << 16)),                // dw2: dim0 hi | dim1 lo
        (int)((ne & 0xFFFFu) << 16),                               // dw3: tile_dim0
        1,                                                         // dw4: tile_dim1=1
        (int)ne,                                                   // dw5: dim0_stride lo
        (int)((ne & 0xFFFFu) << 16),                               // dw6: dim1_stride lo
        (int)(ne >> 16)                                            // dw7: dim1_stride hi
    };
    i32x4 z4 = {0, 0, 0, 0};
#if defined(__clang_major__) && __clang_major__ >= 23
    i32x8 z8 = {0, 0, 0, 0, 0, 0, 0, 0};
    __builtin_amdgcn_tensor_load_to_lds(g0, g1, z4, z4, z8, 0);
#else
    __builtin_amdgcn_tensor_load_to_lds(g0, g1, z4, z4, 0);
#endif
  }

  // Stage code norms into LDS while the TDM copy is in flight.
  for (int k = tid; k < KQ; k += 128) ens[k] = enorm[k];

  // A fragments: 16 chunks of K=4 f32 (ISA 16x4 layout: v0=K0/K2, v1=K1/K3)
  v2f a[16];
  {
    const float* xr = X + (long)(wrow0 + rm) * DQ + hi * 2;
#pragma unroll
    for (int kk = 0; kk < 16; ++kk) a[kk] = *(const v2f*)(xr + kk * 4);
  }

  if (wave == 0) __builtin_amdgcn_s_wait_tensorcnt(0);
  __syncthreads();

  float best[8];
  int bidx[8];
#pragma unroll
  for (int i = 0; i < 8; ++i) { best[i] = 3.4e38f; bidx[i] = 0; }

  for (int ct = 0; ct < 32; ++ct) {  // 32 tiles of 16 codes
    const int kcol = ct * 16 + rm;
    const float* eb = &cb[kcol * CBS + hi * 2];
    v2f b[16];
#pragma unroll
    for (int kk = 0; kk < 16; ++kk) b[kk] = *(const v2f*)(eb + kk * 4);
    v8f acc = {};
#pragma unroll
    for (int kk = 0; kk < 16; ++kk) {
      acc = __builtin_amdgcn_wmma_f32_16x16x4_f32(
          /*neg_a=*/false, a[kk], /*neg_b=*/false, b[kk],
          /*c_mod=*/(short)0, acc, /*reuse_a=*/false, /*reuse_b=*/false);
    }
    const float en = ens[kcol];
#pragma unroll
    for (int i = 0; i < 8; ++i) {
      float score = en - 2.0f * acc[i];  // ||x||^2 constant per row: drop it
      if (score < best[i]) { best[i] = score; bidx[i] = kcol; }
    }
  }

  // Cross-lane argmin within each 16-lane column group (xor 1..8 stays in group)
#pragma unroll
  for (int i = 0; i < 8; ++i) {
#pragma unroll
    for (int off = 8; off >= 1; off >>= 1) {
      float os = __shfl_xor(best[i], off, 32);
      int oi = __shfl_xor(bidx[i], off, 32);
      if (os < best[i] || (os == best[i] && oi < bidx[i])) {
        best[i] = os;
        bidx[i] = oi;
      }
    }
  }
  if (rm == 0) {  // lane0 -> rows 0..7, lane16 -> rows 8..15
#pragma unroll
    for (int i = 0; i < 8; ++i) idx_s[wave * 16 + hi * 8 + i] = bidx[i];
  }
  __syncthreads();

  // Emit indices (int scratch + float output) + histogram
  if (tid < 64) {
    int id = idx_s[tid];
    int g = row0 + tid;
    idx_g[g] = id;
    out_idxf[g] = (float)id;
    atomicAdd(&counts[id], 1);
  }

  // Quantized (straight-through == gathered code, from LDS codebook) + SSE
  float lsum = 0.f;
#pragma unroll 4
  for (int e = tid; e < 64 * DQ; e += 128) {
    int r = e >> 6, d = e & 63;
    int id = idx_s[r];
    long g = (long)(row0 + r) * DQ + d;
    float q = cb[id * CBS + d];
    float x = X[g];
    __builtin_nontemporal_store(q, &out_q[g]);
    float df = q - x;
    lsum += df * df;
  }
  red_s[tid] = lsum;
  __syncthreads();
  for (int s = 64; s > 0; s >>= 1) {
    if (tid < s) red_s[tid] += red_s[tid + s];
    __syncthreads();
  }
  if (tid == 0) atomicAdd(sse_g, red_s[0]);
}

// ---------------- one-hot encodings: 512 MB non-temporal stream -------------
__global__ void vq_encodings(const int* __restrict__ idx,
                             float* __restrict__ enc) {
  long t = (long)blockIdx.x * blockDim.x + threadIdx.x;  // one float2 each
  const long total = (long)NQ * KQ / 2;
  if (t >= total) return;
  int n = (int)(t >> 8);            // 256 float2 per row (K=512)
  int k2 = (int)(t & 255) * 2;
  int id = idx[n];
  v2f v = {0.f, 0.f};
  if (id == k2) v.x = 1.f;
  else if (id == k2 + 1) v.y = 1.f;
  __builtin_nontemporal_store(v, (v2f*)(enc + 2 * t));
}

// ---------------- loss + perplexity ----------------------------------------
__global__ void vq_finalize(const int* __restrict__ counts,
                            const float* __restrict__ sse,
                            float* __restrict__ out_loss,
                            float* __restrict__ out_perp) {
  __shared__ float red[512];
  int k = threadIdx.x;  // 512 threads
  float p = (float)counts[k] / (float)NQ;
  red[k] = p * logf(p + 1e-10f);
  __syncthreads();
  for (int s = 256; s > 0; s >>= 1) {
    if (k < s) red[k] += red[k + s];
    __syncthreads();
  }
  if (k == 0) {
    *out_perp = expf(-red[0]);
    // q_latent_loss == e_latent_loss numerically -> (1 + commitment) * MSE
    *out_loss = 1.25f * (*sse) / (float)((long)NQ * DQ);
  }
}

extern "C" void kernel_launch(void* const* d_in, const int* in_sizes, int n_in,
                              void* d_out, int out_size, void* d_ws,
                              size_t ws_size, hipStream_t stream) {
  (void)in_sizes; (void)n_in; (void)out_size; (void)ws_size;
  const float* X = (const float*)d_in[0];  // [16,16384,64] f32
  const float* E = (const float*)d_in[1];  // [512,64] f32
  float* out = (float*)d_out;

  const long ND = (long)NQ * DQ;   // 16,777,216
  const long NK = (long)NQ * KQ;   // 134,217,728
  float* out_loss = out;                    // [0]
  float* out_q    = out + 1;                // [1 .. 1+ND)
  float* out_perp = out + 1 + ND;           // [1+ND]
  float* out_enc  = out + 2 + ND;           // [2+ND .. 2+ND+NK)  (8B aligned)
  float* out_idxf = out + 2 + ND + NK;      // [.. +NQ)

  float* ws = (float*)d_ws;
  float* enorm = ws;                 // 512 floats
  int* counts  = (int*)(ws + 512);   // 512 ints
  float* sse   = ws + 1024;          // 1 float
  int* idx     = (int*)(ws + 1028);  // NQ ints (~1 MB)

  vq_prep<<<1, 512, 0, stream>>>(E, enorm, counts, sse);
  vq_argmin<<<NQ / 64, 128, 0, stream>>>(X, E, enorm, idx, out_q, out_idxf,
                                         counts, sse);
  vq_encodings<<<(int)((NK / 2) / 256), 256, 0, stream>>>(idx, out_enc);
  vq_finalize<<<1, 512, 0, stream>>>(counts, sse, out_loss, out_perp);
}